// Detection_Framework_37099927503197
// MI455X (gfx1250) — compile-verified
//
#include <hip/hip_runtime.h>
#include <hip/hip_bf16.h>
#include <stdint.h>

#define A_NUM 3
#define C_NUM 2
#define T_NUM 16
#define B_NUM 2
#define GSIDE 64
#define GG (GSIDE*GSIDE*GSIDE)        /* 262144 */
#define DD (A_NUM*GG)                  /* 786432 */
#define NPRED ((size_t)B_NUM*DD*7)     /* 11,010,048 */
#define SCALE_F 4.0f
#define IGN_THR 0.333f
#define PI_F 3.14159265358979f
#define BLK 256

typedef __attribute__((ext_vector_type(2))) float v2f;
typedef __attribute__((ext_vector_type(8))) float v8f;

__device__ __forceinline__ float sigf(float x)  { return 1.0f / (1.0f + __expf(-x)); }
__device__ __forceinline__ float sigfp(float x) { return 1.0f / (1.0f + expf(-x)); }
__device__ __forceinline__ float sl1(float d)   { float ad = fabsf(d); return ad < 1.0f ? 0.5f*d*d : ad - 0.5f; }
__device__ __forceinline__ float bce(float p, float t) {
  p = fminf(fmaxf(p, 1e-12f), 1.0f - 1e-12f);
  return -(t * logf(p) + (1.0f - t) * logf(1.0f - p));
}
__device__ __forceinline__ float sphere_vol(float r) { return (4.0f/3.0f)*PI_F*r*r*r; }

// ---------------------------------------------------------------- init
__global__ void det_init_kernel(unsigned long long* slots) {
  int i = threadIdx.x;
  if (i < B_NUM*T_NUM) slots[i] = 0x00000000FFFFFFFFull; // pack(iou=0.0, idx=0)
}

// ---------------------------------------------------------------- pass 1: pred
__global__ void det_pred_kernel(const float* __restrict__ lo,
                                const float* __restrict__ anchor,
                                float* __restrict__ out) {
  const int idx = blockIdx.x * BLK + threadIdx.x;       // 0 .. B*D-1
  const int b = idx / DD;
  const int d = idx - b * DD;
  const int a = d / GG;
  const int g = d - a * GG;
  const size_t cb = ((size_t)(b*21 + a*7)) * (size_t)GG + (size_t)g;
  const float v0 = lo[cb + 0*(size_t)GG];
  const float v1 = lo[cb + 1*(size_t)GG];
  const float v2 = lo[cb + 2*(size_t)GG];
  const float v3 = lo[cb + 3*(size_t)GG];
  const float v4 = lo[cb + 4*(size_t)GG];
  const float v5 = lo[cb + 5*(size_t)GG];
  const float v6 = lo[cb + 6*(size_t)GG];
  const int ix = g & 63, iy = (g >> 6) & 63, iz = g >> 12;
  const float px = (sigf(v0) + (float)ix) * SCALE_F;
  const float py = (sigf(v1) + (float)iy) * SCALE_F;
  const float pz = (sigf(v2) + (float)iz) * SCALE_F;
  const float pr = __expf(v3) * anchor[a];
  const float cf = sigf(v4);
  const float mx = fmaxf(v5, v6);
  const float e0 = __expf(v5 - mx), e1 = __expf(v6 - mx);
  const float inv = 1.0f / (e0 + e1);
  const size_t ob = (size_t)idx * 7;
  out[ob + 0] = px; out[ob + 1] = py; out[ob + 2] = pz; out[ob + 3] = pr;
  out[ob + 4] = cf; out[ob + 5] = e0 * inv; out[ob + 6] = e1 * inv;
}

// ---------------------------------------------------------------- pass 2: WMMA d^2 + IoU + argmax + bg sum
__global__ void det_iou_kernel(const float* __restrict__ pred,
                               const float* __restrict__ targets,
                               unsigned long long* __restrict__ slots,
                               float* __restrict__ partials) {
  __shared__ float tg[T_NUM][8];
  __shared__ float red[BLK];
  const int tid = threadIdx.x;
  const int blocksPerB = DD / BLK;                       // 3072
  const int b  = blockIdx.x / blocksPerB;
  const int d0 = (blockIdx.x - b * blocksPerB) * BLK;
  if (tid < T_NUM * 8) tg[tid >> 3][tid & 7] = targets[(size_t)b * T_NUM * 8 + tid];
  __syncthreads();

  const int lane = tid & 31;
  const int dloc = d0 + tid;                             // this lane's prediction
  const size_t pb = ((size_t)b * DD + dloc) * 7;
  const float bx = pred[pb+0], by = pred[pb+1], bz = pred[pb+2], bw = pred[pb+3];
  const float pconf = pred[pb+4];

  // A matrix (16x4 f32): row M = target, K = {-2tx,-2ty,-2tz,|t|^2}
  // layout: VGPR0 lanes0-15 K=0 / lanes16-31 K=2 ; VGPR1 K=1 / K=3
  const int  m  = lane & 15;
  const bool hi = lane >= 16;
  const float tx = tg[m][0], ty = tg[m][1], tz = tg[m][2];
  const float dott = __fmaf_rn(tx, tx, __fmaf_rn(ty, ty, tz * tz));
  v2f Am; Am.x = hi ? (-2.0f*tz) : (-2.0f*tx);
          Am.y = hi ? dott       : (-2.0f*ty);
  const float pp = __fmaf_rn(bx, bx, __fmaf_rn(by, by, bz * bz));

  unsigned long long ig_ballot[2];
  const int wbase = d0 + (tid & ~31);

  #pragma unroll
  for (int tile = 0; tile < 2; ++tile) {
    const int src = tile*16 + m;                         // prediction column N=m of this tile
    const float sx  = __shfl(bx, src);
    const float sy  = __shfl(by, src);
    const float sz  = __shfl(bz, src);
    const float sw  = __shfl(bw, src);
    const float spp = __shfl(pp, src);
    // B matrix (4x16 f32): VGPR0 rows K=0/K=2, VGPR1 rows K=1/K=3 (mirror of A)
    v2f Bm; Bm.x = hi ? sz   : sx;
            Bm.y = hi ? 1.0f : sy;
    v8f C = { spp, spp, spp, spp, spp, spp, spp, spp };  // |p|^2 per column
    // D2[k] = squared distance of target ((hi?8:0)+k) to prediction column m
    v8f D2 = __builtin_amdgcn_wmma_f32_16x16x4_f32(false, Am, false, Bm,
                                                   (short)0, C, false, false);
    bool lig = false;
    const int dg = wbase + src;                          // global pred index of column
    #pragma unroll
    for (int k = 0; k < 8; ++k) {
      const int kk = (hi ? 8 : 0) + k;
      const float r1  = tg[kk][3];
      const bool  val = tg[kk][4] > 0.5f;
      const float d2  = D2[k];
      const float rsum = r1 + sw;
      if (val && d2 < rsum * rsum) {                     // else IoU == 0: pruned
        const float dd   = sqrtf(d2 + 1e-12f);
        const float rdif = fabsf(r1 - sw);
        float inter;
        if (dd <= rdif) {
          inter = sphere_vol(fminf(r1, sw));
        } else {
          const float t1 = rsum - dd;
          inter = PI_F * t1 * t1 * (d2 + 2.0f*dd*rsum - 3.0f*rdif*rdif) / (12.0f*dd);
        }
        const float iou = inter / (sphere_vol(r1) + sphere_vol(sw) - inter + 1e-9f);
        if (iou > IGN_THR) lig = true;
        if (iou > 0.0f) {
          const unsigned long long pk =
              ((unsigned long long)__float_as_uint(iou) << 32) |
              (unsigned long long)(0xFFFFFFFFu - (unsigned)dg);  // smallest idx wins ties
          atomicMax(&slots[b * T_NUM + kk], pk);
        }
      }
    }
    ig_ballot[tile] = __ballot(lig);
  }

  // my own prediction's ignore flag: bits (m) and (m+16) of my tile's ballot
  const unsigned long long bal = ig_ballot[lane >> 4];
  const int mb = lane & 15;
  const bool ignored = (((bal >> mb) | (bal >> (mb + 16))) & 1ull) != 0;
  const float bce0 = -__logf(fmaxf(1.0f - pconf, 1e-12f));
  red[tid] = ignored ? 0.0f : bce0;
  __syncthreads();
  #pragma unroll
  for (int s = BLK/2; s > 0; s >>= 1) {
    if (tid < s) red[tid] += red[tid + s];
    __syncthreads();
  }
  if (tid == 0) partials[blockIdx.x] = red[0];
}

// ---------------------------------------------------------------- pass 3: finish (1 block)
__global__ void det_finish_kernel(const float* __restrict__ lo,
                                  const float* __restrict__ targets,
                                  const float* __restrict__ anchor,
                                  const unsigned long long* __restrict__ slots,
                                  const float* __restrict__ partials, int npart,
                                  float* __restrict__ out) {
  __shared__ float s_bb[32], s_cls[32], s_bce1[32], s_bce0n[32];
  __shared__ int   s_d[32], s_val[32];
  __shared__ float red[64];
  const int tid = threadIdx.x;

  // deterministic fixed-order reduction of background-BCE partials
  float acc = 0.0f;
  for (int i = tid; i < npart; i += 64) acc += partials[i];
  red[tid] = acc;
  __syncthreads();
  #pragma unroll
  for (int s = 32; s > 0; s >>= 1) { if (tid < s) red[tid] += red[tid + s]; __syncthreads(); }
  const float S = red[0];

  if (tid < 32) {
    const int b = tid >> 4, t = tid & 15;
    const float* tgt = &targets[(size_t)(b * T_NUM + t) * 8];
    const bool val = tgt[4] > 0.5f;
    const unsigned long long v = slots[tid];
    const int ds = (int)(0xFFFFFFFFu - (unsigned)(v & 0xFFFFFFFFull));
    s_d[tid] = ds; s_val[tid] = val ? 1 : 0;
    float lb = 0.0f, lc = 0.0f, b1 = 0.0f, b0n = 0.0f;
    if (val) {
      const int a = ds / GG, g = ds - a * GG;
      float raw[7];
      #pragma unroll
      for (int c = 0; c < 7; ++c)
        raw[c] = lo[((size_t)(b*21 + a*7 + c)) * (size_t)GG + (size_t)g];
      const float xs = sigfp(raw[0]);
      const float ys = sigfp(raw[1]);
      const float zs = sigfp(raw[2]);
      const float rr = raw[3];
      const float cf = sigfp(raw[4]);
      const float mx = fmaxf(raw[5], raw[6]);
      const float e0 = expf(raw[5]-mx), e1 = expf(raw[6]-mx);
      const float c0 = e0/(e0+e1), c1 = e1/(e0+e1);
      const float q0 = tgt[0]/SCALE_F, q1 = tgt[1]/SCALE_F, q2 = tgt[2]/SCALE_F;
      const float tb0 = q0 - truncf(q0), tb1 = q1 - truncf(q1), tb2 = q2 - truncf(q2);
      const float tb3 = logf(tgt[3] / anchor[a]);
      lb = sl1(xs-tb0) + sl1(ys-tb1) + sl1(zs-tb2) + sl1(rr-tb3);
      lc = bce(c0, tgt[5]) + bce(c1, tgt[6]);
      const float pcl = fminf(fmaxf(cf, 1e-12f), 1.0f - 1e-12f);
      b1 = -logf(pcl);
      const float bce0w = -logf(1.0f - pcl);
      // winner's ignore flag: IoU against all valid targets of batch b
      const int ix = g & 63, iy = (g >> 6) & 63, iz = g >> 12;
      const float pxw = (xs + ix) * SCALE_F, pyw = (ys + iy) * SCALE_F,
                  pzw = (zs + iz) * SCALE_F, prw = expf(rr) * anchor[a];
      bool ign = false;
      for (int u = 0; u < T_NUM; ++u) {
        const float* tu = &targets[(size_t)(b * T_NUM + u) * 8];
        if (!(tu[4] > 0.5f)) continue;
        const float dx = tu[0]-pxw, dy = tu[1]-pyw, dz = tu[2]-pzw;
        const float d2 = dx*dx + dy*dy + dz*dz;
        const float r1 = tu[3], rsum = r1 + prw;
        if (d2 >= rsum * rsum) continue;
        const float dd = sqrtf(d2 + 1e-12f);
        const float rdif = fabsf(r1 - prw);
        float inter;
        if (dd <= rdif) inter = sphere_vol(fminf(r1, prw));
        else { const float t1 = rsum - dd;
               inter = PI_F*t1*t1*(d2 + 2.0f*dd*rsum - 3.0f*rdif*rdif)/(12.0f*dd); }
        const float iou = inter / (sphere_vol(r1) + sphere_vol(prw) - inter + 1e-9f);
        if (iou > IGN_THR) { ign = true; break; }
      }
      b0n = ign ? 0.0f : bce0w;
    }
    s_bb[tid] = lb; s_cls[tid] = lc; s_bce1[tid] = b1; s_bce0n[tid] = b0n;
  }
  __syncthreads();

  if (tid == 0) {
    float lbs = 0.0f, lcs = 0.0f, obj = 0.0f, corr = 0.0f;
    for (int b = 0; b < B_NUM; ++b)
      for (int t = 0; t < T_NUM; ++t) {
        const int i = b * T_NUM + t;
        if (!s_val[i]) continue;
        lbs += s_bb[i]; lcs += s_cls[i];
        bool dup = false;
        for (int u = 0; u < t; ++u) {
          const int j = b * T_NUM + u;
          if (s_val[j] && s_d[j] == s_d[i]) { dup = true; break; }
        }
        if (!dup) { obj += s_bce1[i]; corr += s_bce0n[i]; }
      }
    const float loss_bbox = 5.0f * lbs / (float)B_NUM;
    const float loss_cls  = 5.0f * lcs / (float)B_NUM;
    const float loss_conf = (1.0f * obj + 0.1f * (S - corr)) / (float)B_NUM;
    out[NPRED + 0] = loss_bbox + loss_conf + loss_cls;
    out[NPRED + 1] = loss_conf;
    out[NPRED + 2] = loss_cls;
  }
}

// ---------------------------------------------------------------- launch
extern "C" void kernel_launch(void* const* d_in, const int* in_sizes, int n_in,
                              void* d_out, int out_size, void* d_ws, size_t ws_size,
                              hipStream_t stream) {
  (void)in_sizes; (void)n_in; (void)out_size; (void)ws_size;
  const float* lo      = (const float*)d_in[0];   // layer_out (2,21,64,64,64)
  const float* targets = (const float*)d_in[1];   // (2,16,8)
  const float* anchor  = (const float*)d_in[2];   // (3,3,1); anc0 = flat[0..2]
  float* out = (float*)d_out;

  unsigned long long* slots = (unsigned long long*)d_ws;
  float* partials = (float*)((char*)d_ws + 64 * sizeof(unsigned long long));

  det_init_kernel<<<1, 64, 0, stream>>>(slots);

  const int ntot = B_NUM * DD;                    // 1,572,864 threads
  det_pred_kernel<<<ntot / BLK, BLK, 0, stream>>>(lo, anchor, out);

  const int nblk2 = B_NUM * DD / BLK;             // 6144 blocks
  det_iou_kernel<<<nblk2, BLK, 0, stream>>>(out, targets, slots, partials);

  det_finish_kernel<<<1, 64, 0, stream>>>(lo, targets, anchor, slots, partials, nblk2, out);
}